// SoftDTWLoss_81423989997803
// MI455X (gfx1250) — compile-verified
//
#include <hip/hip_runtime.h>

// Soft-DTW (gamma = 1) forward + backward for B=16, N=M=1024 on gfx1250.
// Output layout (f32): score[16] | D[16*1024*1024] | R[16*1025*1025] | E[16*1024*1024]

#define BIGV 1.0e10f

typedef float v2f __attribute__((ext_vector_type(2)));
typedef float v8f __attribute__((ext_vector_type(8)));

constexpr int BB = 16;
constexpr int NN = 1024;
constexpr int MM = 1024;
constexpr size_t D_OFF = 16;
constexpr size_t R_OFF = D_OFF + (size_t)BB * NN * MM;
constexpr size_t E_OFF = R_OFF + (size_t)BB * (NN + 1) * (MM + 1);

#if defined(__has_builtin)
#if __has_builtin(__builtin_amdgcn_wmma_f32_16x16x4_f32)
#define HAVE_WMMA_F32_16X16X4 1
#endif
#endif

// ---------------------------------------------------------------------------
// Kernel 1: D[b,i,j] = (x[b,i] - y[j])^2 via WMMA rank-1 outer product.
// D = x^2 - 2*(x*y) + y^2 ; the x*y outer product for a 16x16 tile comes from
// V_WMMA_F32_16X16X4_F32 with x in A's K=0 column and y in B's K=0 row.
// One wave owns a 16-row strip and sweeps all 64 column tiles.
// ---------------------------------------------------------------------------
__global__ __launch_bounds__(256) void sdtw_d_wmma_kernel(
    const float* __restrict__ x, const float* __restrict__ y,
    float* __restrict__ out) {
  const int lane = threadIdx.x & 31;
  const int wv = threadIdx.x >> 5;
  const int b = blockIdx.x;
  const int i0 = (blockIdx.y * 8 + wv) * 16;  // 0..1008
  const float* xb = x + (size_t)b * NN;
  float* Db = out + D_OFF + (size_t)b * NN * MM;

  // A operand: 16x4 f32. lanes 0-15 hold K=0 (V0) / K=1 (V1), lanes 16-31 hold K=2/3.
  v2f A;
  A.x = (lane < 16) ? xb[i0 + lane] : 0.0f;  // K=0 column = x
  A.y = 0.0f;

  // C/D layout: lanes 0-15 -> M=r, lanes 16-31 -> M=8+r ; N = lane&15.
  const int mbase = (lane >= 16) ? 8 : 0;
  const int n = lane & 15;
  float xr[8], x2r[8];
#pragma unroll
  for (int r = 0; r < 8; ++r) {
    xr[r] = xb[i0 + mbase + r];
    x2r[r] = xr[r] * xr[r];
  }

  for (int j0 = 0; j0 < MM; j0 += 16) {
    // B operand: 4x16 f32. lanes 0-15 hold K=0 (V0) / K=1 (V1), lanes 16-31 K=2/3.
    v2f Bm;
    Bm.x = (lane < 16) ? y[j0 + lane] : 0.0f;  // K=0 row = y
    Bm.y = 0.0f;
    const float yv = y[j0 + n];
    const float y2 = yv * yv;

    v8f P;
#if defined(HAVE_WMMA_F32_16X16X4)
    v8f C = {0.f, 0.f, 0.f, 0.f, 0.f, 0.f, 0.f, 0.f};
    P = __builtin_amdgcn_wmma_f32_16x16x4_f32(false, A, false, Bm, (short)0, C,
                                              false, false);
#else
#pragma unroll
    for (int r = 0; r < 8; ++r) P[r] = xr[r] * yv;
#endif

#pragma unroll
    for (int r = 0; r < 8; ++r) {
      const float dv = (x2r[r] - 2.0f * P[r]) + y2;
      Db[(size_t)(i0 + mbase + r) * MM + (j0 + n)] = dv;
    }
  }
}

// ---------------------------------------------------------------------------
// Kernel 2: forward DP. One block (1024 threads / 32 waves) per batch.
// Anti-diagonal wavefront, 3 rotating diagonals in LDS, D recomputed exactly.
// ---------------------------------------------------------------------------
__global__ __launch_bounds__(1024) void sdtw_forward_kernel(
    const float* __restrict__ x, const float* __restrict__ y,
    float* __restrict__ out) {
  __shared__ float sx[NN];
  __shared__ float sy[MM];
  __shared__ float rbuf[3][NN + 1];

  const int tid = threadIdx.x;
  const int b = blockIdx.x;
  float* Rb = out + R_OFF + (size_t)b * (NN + 1) * (MM + 1);
  float* score = out;  // first 16 floats

  sx[tid] = x[(size_t)b * NN + tid];
  sy[tid] = y[tid];

  // diag 0: R[0][0]=0 (rest BIG placeholders); diag 1: all BIG.
  rbuf[0][tid] = (tid == 0) ? 0.0f : BIGV;
  rbuf[1][tid] = BIGV;
  if (tid == 0) {
    rbuf[0][NN] = BIGV;
    rbuf[1][NN] = BIGV;
  }
  // Boundary of R: row 0 and column 0.
  Rb[tid + 1] = BIGV;                        // R[0][1..1024]
  Rb[(size_t)(tid + 1) * (MM + 1)] = BIGV;   // R[1..1024][0]
  if (tid == 0) Rb[0] = 0.0f;

  for (int d = 2; d <= NN + MM; ++d) {
    __syncthreads();
    float* cur = rbuf[d % 3];
    const float* p1 = rbuf[(d + 2) % 3];  // diag d-1
    const float* p2 = rbuf[(d + 1) % 3];  // diag d-2
    const int i = tid + 1;
    const int j = d - i;
    float val = BIGV;
    if (j >= 1 && j <= MM) {
      const float a = p2[i - 1];   // R[i-1][j-1]
      const float bv = p1[i - 1];  // R[i-1][j]
      const float c = p1[i];       // R[i][j-1]
      const float m = fminf(a, fminf(bv, c));
      const float s = __expf(m - a) + __expf(m - bv) + __expf(m - c);
      const float sm = m - __logf(s);
      const float t = sx[i - 1] - sy[j - 1];
      val = t * t + sm;
      Rb[(size_t)i * (MM + 1) + j] = val;
      if (d == NN + MM && i == NN) score[b] = val;
    }
    cur[i] = val;
    if (tid == 0) cur[0] = BIGV;
  }
}

// ---------------------------------------------------------------------------
// Kernel 3: backward DP (vjp adjoint). Reverse wavefront; R read from global
// (L2-resident) one diagonal per step; exp args are <= 0 by construction.
// ---------------------------------------------------------------------------
__global__ __launch_bounds__(1024) void sdtw_backward_kernel(
    const float* __restrict__ x, const float* __restrict__ y,
    float* __restrict__ out) {
  __shared__ float sx[NN];
  __shared__ float sy[MM];
  __shared__ float rr[3][NN + 1];
  __shared__ float ee[3][NN + 1];

  const int tid = threadIdx.x;
  const int b = blockIdx.x;
  const float* Rb = out + R_OFF + (size_t)b * (NN + 1) * (MM + 1);
  float* Eb = out + E_OFF + (size_t)b * NN * MM;

  sx[tid] = x[(size_t)b * NN + tid];
  sy[tid] = y[tid];
  for (int k = 0; k < 3; ++k) {
    rr[k][tid] = BIGV;
    ee[k][tid] = 0.0f;
  }
  if (tid == 0) {
    for (int k = 0; k < 3; ++k) {
      rr[k][NN] = BIGV;
      ee[k][NN] = 0.0f;
    }
    // Seed: diag d=2048 (index 2048%3==2) holds only cell (N,M): E=1.
    ee[2][NN] = 1.0f;
    rr[2][NN] = Rb[(size_t)NN * (MM + 1) + MM];
    Eb[(size_t)(NN - 1) * MM + (MM - 1)] = 1.0f;
  }

  for (int d = NN + MM - 1; d >= 2; --d) {
    __syncthreads();
    const int i = tid + 1;
    const int j = d - i;
    float* rcur = rr[d % 3];
    float* ecur = ee[d % 3];
    const float* r1 = rr[(d + 1) % 3];
    const float* e1 = ee[(d + 1) % 3];
    const float* r2 = rr[(d + 2) % 3];
    const float* e2 = ee[(d + 2) % 3];

    float eij = 0.0f;
    float rij = BIGV;
    if (j >= 1 && j <= MM) {
      rij = Rb[(size_t)i * (MM + 1) + j];
      // Prefetch next diagonal's R element for this row (gfx1250 global_prefetch_b8).
      __builtin_prefetch(Rb + (size_t)i * (MM + 1) + (j - 1), 0, 1);
      float acc = 0.0f;
      float t;
      if (i < NN && j < MM) {  // child (i+1, j+1), D0[i][j]
        t = sx[i] - sy[j];
        acc += e2[i + 1] * __expf(r2[i + 1] - rij - t * t);
      }
      if (i < NN) {  // child (i+1, j), D0[i][j-1]
        t = sx[i] - sy[j - 1];
        acc += e1[i + 1] * __expf(r1[i + 1] - rij - t * t);
      }
      if (j < MM) {  // child (i, j+1), D0[i-1][j]
        t = sx[i - 1] - sy[j];
        acc += e1[i] * __expf(r1[i] - rij - t * t);
      }
      eij = acc;
      Eb[(size_t)(i - 1) * MM + (j - 1)] = eij;
    }
    rcur[i] = rij;
    ecur[i] = eij;
    if (tid == 0) {
      rcur[0] = BIGV;
      ecur[0] = 0.0f;
    }
  }
}

// ---------------------------------------------------------------------------
extern "C" void kernel_launch(void* const* d_in, const int* in_sizes, int n_in,
                              void* d_out, int out_size, void* d_ws,
                              size_t ws_size, hipStream_t stream) {
  (void)in_sizes; (void)n_in; (void)out_size; (void)d_ws; (void)ws_size;
  const float* x = (const float*)d_in[0];  // [16, 1024]
  const float* y = (const float*)d_in[1];  // [1024]
  float* out = (float*)d_out;

  dim3 gD(BB, 8, 1);
  sdtw_d_wmma_kernel<<<gD, dim3(256, 1, 1), 0, stream>>>(x, y, out);
  sdtw_forward_kernel<<<dim3(BB, 1, 1), dim3(1024, 1, 1), 0, stream>>>(x, y, out);
  sdtw_backward_kernel<<<dim3(BB, 1, 1), dim3(1024, 1, 1), 0, stream>>>(x, y, out);
}